// RoleTransitionPredictor_41970420418031
// MI455X (gfx1250) — compile-verified
//
#include <hip/hip_runtime.h>
#include <math.h>

// ---------------- problem constants ----------------
#define N_NODES 100000
#define N_EDGES 1600000
#define FDIM    128
#define HDIM    128
#define BATCH   16384
#define NROLE   5
#define CLIPV   10.0f
#define APAD    132   // padded LDS row stride (floats): bank stride 4 -> conflict-free fragment reads

typedef __attribute__((ext_vector_type(2))) float v2f;
typedef __attribute__((ext_vector_type(8))) float v8f;

__device__ __forceinline__ float clipf(float v) { return fminf(CLIPV, fmaxf(-CLIPV, v)); }
__device__ __forceinline__ float sigf(float v)  { return 1.0f / (1.0f + expf(-v)); }

// fp32 WMMA: D = A(16x4) * B(4x16) + C ; wave32, one tile per wave
__device__ __forceinline__ v8f wmma_f32(v2f a, v2f b, v8f c) {
    return __builtin_amdgcn_wmma_f32_16x16x4_f32(false, a, false, b, (short)0, c, false, false);
}

// CDNA5 async global->LDS copy, 16 bytes per lane; tracked by ASYNCcnt
__device__ __forceinline__ void async_ld_b128(void* lds_dst, const float* gsrc) {
    unsigned lds_off = (unsigned)(reinterpret_cast<uintptr_t>(lds_dst)); // flat shared: low 32 bits = LDS addr
    asm volatile("global_load_async_to_lds_b128 %0, %1, off"
                 :: "v"(lds_off), "v"(gsrc)
                 : "memory");
}
__device__ __forceinline__ void wait_async_all() {
    asm volatile("s_wait_asynccnt 0x0" ::: "memory");
}

__device__ __forceinline__ void atomAddF(float* p, float v) {
    unsafeAtomicAdd(p, v);  // native global_atomic_add_f32
}

// ---------------- utility kernels ----------------
__global__ void zero_kernel(float* __restrict__ p, int n) {
    int i = blockIdx.x * blockDim.x + threadIdx.x;
    int stride = gridDim.x * blockDim.x;
    for (; i < n; i += stride) p[i] = 0.0f;
}

__global__ void degree_kernel(const int* __restrict__ dst, float* __restrict__ deg, int e) {
    int i = blockIdx.x * blockDim.x + threadIdx.x;
    if (i < e) atomAddF(&deg[dst[i]], 1.0f);
}

// one thread handles 4 consecutive features of one edge (32 threads / edge)
__global__ void scatter_kernel(const float* __restrict__ hin, const int* __restrict__ src,
                               const int* __restrict__ dst, float* __restrict__ msg, int e) {
    int t = blockIdx.x * blockDim.x + threadIdx.x;
    int eidx = t >> 5;
    if (eidx >= e) return;
    int f0 = (t & 31) * 4;
    int s = src[eidx], d = dst[eidx];
    const float4 v = *(const float4*)(hin + (size_t)s * FDIM + f0);
    float* mp = msg + (size_t)d * FDIM + f0;
    atomAddF(mp + 0, v.x);
    atomAddF(mp + 1, v.y);
    atomAddF(mp + 2, v.z);
    atomAddF(mp + 3, v.w);
}

// ---------------- SAGE dense layer: out = act( (msg/deg) @ wl^T + bl + hin @ wr^T ) ----------------
// block = 256 threads (8 waves), each wave does one 16x16 output tile; grid.x = Nrows/16.
// The 16 A rows (msg + hin) are shared by all 8 waves -> stage once in LDS via async copy.
__global__ void sage_gemm_kernel(const float* __restrict__ msg, const float* __restrict__ deg,
                                 const float* __restrict__ hin,
                                 const float* __restrict__ wl, const float* __restrict__ wr,
                                 const float* __restrict__ bl,
                                 float* __restrict__ out, int do_relu) {
    __shared__ float sMsg[16 * APAD];
    __shared__ float sHin[16 * APAD];
    const int wave = threadIdx.x >> 5;
    const int lane = threadIdx.x & 31;
    const int half = lane >> 4;
    const int l16  = lane & 15;
    const int m0 = blockIdx.x * 16;
    const int n0 = wave * 16;

    // cooperative async fill: 16 rows x 32 four-float groups per matrix = 512 groups; 256 thr x 2 passes
#pragma unroll
    for (int pass = 0; pass < 2; pass++) {
        const int gl = threadIdx.x + pass * 256;   // 0..511
        const int r  = gl >> 5;                    // row 0..15
        const int g  = (gl & 31) * 4;              // col 0,4,...,124
        async_ld_b128(&sMsg[r * APAD + g], msg + (size_t)(m0 + r) * FDIM + g);
        async_ld_b128(&sHin[r * APAD + g], hin + (size_t)(m0 + r) * FDIM + g);
    }
    wait_async_all();
    __syncthreads();

    const float inv = 1.0f / fmaxf(deg[m0 + l16], 1.0f);
    const float* __restrict__ wlRow = wl + (size_t)(n0 + l16) * FDIM;
    const float* __restrict__ wrRow = wr + (size_t)(n0 + l16) * FDIM;
    const float* sMsgRow = &sMsg[l16 * APAD];
    const float* sHinRow = &sHin[l16 * APAD];

    v8f acc = {};
#pragma unroll 4
    for (int kb = 0; kb < FDIM; kb += 4) {
        const int k = kb + 2 * half;
        v2f a0 = *(const v2f*)(sMsgRow + k);
        a0.x *= inv; a0.y *= inv;
        v2f b0; b0.x = wlRow[k]; b0.y = wlRow[k + 1];
        acc = wmma_f32(a0, b0, acc);
        v2f a1 = *(const v2f*)(sHinRow + k);
        v2f b1; b1.x = wrRow[k]; b1.y = wrRow[k + 1];
        acc = wmma_f32(a1, b1, acc);
    }
    const float bias = bl[n0 + l16];
#pragma unroll
    for (int r = 0; r < 8; r++) {
        const int m = m0 + r + 8 * half;
        float v = acc[r] + bias;
        if (do_relu) v = fmaxf(v, 0.0f);
        out[(size_t)m * HDIM + n0 + l16] = v;
    }
}

// layer-2 SAGE evaluated only at gathered user rows; writes clipped ue [BATCH, HDIM]
__global__ void sage_gemm_gather_kernel(const float* __restrict__ msg, const float* __restrict__ deg,
                                        const float* __restrict__ h, const int* __restrict__ uids,
                                        const float* __restrict__ wl, const float* __restrict__ wr,
                                        const float* __restrict__ bl, float* __restrict__ ue) {
    __shared__ float sMsg[16 * APAD];
    __shared__ float sHin[16 * APAD];
    const int wave = threadIdx.x >> 5;
    const int lane = threadIdx.x & 31;
    const int half = lane >> 4;
    const int l16  = lane & 15;
    const int m0 = blockIdx.x * 16;
    const int n0 = wave * 16;

#pragma unroll
    for (int pass = 0; pass < 2; pass++) {
        const int gl = threadIdx.x + pass * 256;
        const int r  = gl >> 5;
        const int g  = (gl & 31) * 4;
        const int grow = uids[m0 + r];
        async_ld_b128(&sMsg[r * APAD + g], msg + (size_t)grow * FDIM + g);
        async_ld_b128(&sHin[r * APAD + g], h + (size_t)grow * FDIM + g);
    }
    wait_async_all();
    __syncthreads();

    const float inv = 1.0f / fmaxf(deg[uids[m0 + l16]], 1.0f);
    const float* __restrict__ wlRow = wl + (size_t)(n0 + l16) * HDIM;
    const float* __restrict__ wrRow = wr + (size_t)(n0 + l16) * HDIM;
    const float* sMsgRow = &sMsg[l16 * APAD];
    const float* sHinRow = &sHin[l16 * APAD];

    v8f acc = {};
#pragma unroll 4
    for (int kb = 0; kb < HDIM; kb += 4) {
        const int k = kb + 2 * half;
        v2f a0 = *(const v2f*)(sMsgRow + k);
        a0.x *= inv; a0.y *= inv;
        v2f b0; b0.x = wlRow[k]; b0.y = wlRow[k + 1];
        acc = wmma_f32(a0, b0, acc);
        v2f a1 = *(const v2f*)(sHinRow + k);
        v2f b1; b1.x = wrRow[k]; b1.y = wrRow[k + 1];
        acc = wmma_f32(a1, b1, acc);
    }
    const float bias = bl[n0 + l16];
#pragma unroll
    for (int r = 0; r < 8; r++) {
        const int m = m0 + r + 8 * half;
        ue[(size_t)m * HDIM + n0 + l16] = clipf(acc[r] + bias);
    }
}

// LSTM gates: gates[m,n] = sum_k combined[m,k]*w_ih[n,k] + b_ih[n]+b_hh[n]+w_ih[n,256+role[m]]
// combined = [ue(128) | clip(x[uid])(128)]; onehot folded into epilogue. grid (BATCH/16, 4), block 256
__global__ void lstm_gates_kernel(const float* __restrict__ ue, const float* __restrict__ x,
                                  const int* __restrict__ uids, const int* __restrict__ roles,
                                  const float* __restrict__ w_ih, const float* __restrict__ b_ih,
                                  const float* __restrict__ b_hh, float* __restrict__ gates) {
    const int wave = threadIdx.x >> 5;
    const int lane = threadIdx.x & 31;
    const int half = lane >> 4;
    const int l16  = lane & 15;
    const int m0 = blockIdx.x * 16;
    const int n0 = (blockIdx.y * 8 + wave) * 16;

    const int rowA = m0 + l16;
    const int uid  = uids[rowA];
    const float* __restrict__ ueRow = ue + (size_t)rowA * HDIM;
    const float* __restrict__ xRow  = x  + (size_t)uid * FDIM;
    const float* __restrict__ wRow  = w_ih + (size_t)(n0 + l16) * (2 * HDIM + NROLE);

    v8f acc = {};
#pragma unroll 4
    for (int kb = 0; kb < 2 * HDIM; kb += 4) {
        const int k = kb + 2 * half;
        v2f a;
        if (k < HDIM) { a.x = ueRow[k];              a.y = ueRow[k + 1]; }
        else          { a.x = clipf(xRow[k - HDIM]); a.y = clipf(xRow[k - HDIM + 1]); }
        v2f b; b.x = wRow[k]; b.y = wRow[k + 1];
        acc = wmma_f32(a, b, acc);
    }
    const int n = n0 + l16;
    const float bias = b_ih[n] + b_hh[n];
#pragma unroll
    for (int r = 0; r < 8; r++) {
        const int m = m0 + r + 8 * half;
        gates[(size_t)m * (4 * HDIM) + n] = acc[r] + bias + wRow[2 * HDIM + roles[m]];
    }
}

// lstm_out = clip( sigmoid(o) * tanh( sigmoid(i)*tanh(g) ) ); f-gate unused (c0 = 0)
__global__ void lstm_act_kernel(const float* __restrict__ gates, float* __restrict__ lstm) {
    int t = blockIdx.x * blockDim.x + threadIdx.x;
    if (t >= BATCH * HDIM) return;
    int b = t >> 7, j = t & (HDIM - 1);
    const float* gRow = gates + (size_t)b * (4 * HDIM);
    float i_ = gRow[j], g_ = gRow[2 * HDIM + j], o_ = gRow[3 * HDIM + j];
    float c = sigf(i_) * tanhf(g_);
    lstm[t] = clipf(sigf(o_) * tanhf(c));
}

// z0 = relu([ue|lstm] @ wc0^T + bc0); K=256, 8 n-tiles, block 256
__global__ void mlp0_kernel(const float* __restrict__ ue, const float* __restrict__ lstm,
                            const float* __restrict__ wc0, const float* __restrict__ bc0,
                            float* __restrict__ z0) {
    const int wave = threadIdx.x >> 5;
    const int lane = threadIdx.x & 31;
    const int half = lane >> 4;
    const int l16  = lane & 15;
    const int m0 = blockIdx.x * 16;
    const int n0 = wave * 16;

    const int rowA = m0 + l16;
    const float* __restrict__ a0Row = ue   + (size_t)rowA * HDIM;
    const float* __restrict__ a1Row = lstm + (size_t)rowA * HDIM;
    const float* __restrict__ wRow  = wc0  + (size_t)(n0 + l16) * (2 * HDIM);

    v8f acc = {};
#pragma unroll 4
    for (int kb = 0; kb < 2 * HDIM; kb += 4) {
        const int k = kb + 2 * half;
        v2f a;
        if (k < HDIM) { a.x = a0Row[k];        a.y = a0Row[k + 1]; }
        else          { a.x = a1Row[k - HDIM]; a.y = a1Row[k - HDIM + 1]; }
        v2f b; b.x = wRow[k]; b.y = wRow[k + 1];
        acc = wmma_f32(a, b, acc);
    }
    const float bias = bc0[n0 + l16];
#pragma unroll
    for (int r = 0; r < 8; r++) {
        const int m = m0 + r + 8 * half;
        z0[(size_t)m * HDIM + n0 + l16] = fmaxf(acc[r] + bias, 0.0f);
    }
}

// z1 = relu(z0 @ wc1^T + bc1); K=128, 64 out cols = 4 n-tiles, block 128
__global__ void mlp1_kernel(const float* __restrict__ z0, const float* __restrict__ wc1,
                            const float* __restrict__ bc1, float* __restrict__ z1) {
    const int wave = threadIdx.x >> 5;
    const int lane = threadIdx.x & 31;
    const int half = lane >> 4;
    const int l16  = lane & 15;
    const int m0 = blockIdx.x * 16;
    const int n0 = wave * 16;

    const int rowA = m0 + l16;
    const float* __restrict__ aRow = z0  + (size_t)rowA * HDIM;
    const float* __restrict__ wRow = wc1 + (size_t)(n0 + l16) * HDIM;

    v8f acc = {};
#pragma unroll 4
    for (int kb = 0; kb < HDIM; kb += 4) {
        const int k = kb + 2 * half;
        v2f a; a.x = aRow[k]; a.y = aRow[k + 1];
        v2f b; b.x = wRow[k]; b.y = wRow[k + 1];
        acc = wmma_f32(a, b, acc);
    }
    const float bias = bc1[n0 + l16];
#pragma unroll
    for (int r = 0; r < 8; r++) {
        const int m = m0 + r + 8 * half;
        z1[(size_t)m * (HDIM / 2) + n0 + l16] = fmaxf(acc[r] + bias, 0.0f);
    }
}

// out = z1 @ wc2^T + bc2; K=64, only 5 valid output cols (mask N tile), block 32
__global__ void mlp2_kernel(const float* __restrict__ z1, const float* __restrict__ wc2,
                            const float* __restrict__ bc2, float* __restrict__ out) {
    const int lane = threadIdx.x & 31;
    const int half = lane >> 4;
    const int l16  = lane & 15;
    const int m0 = blockIdx.x * 16;

    const int rowA = m0 + l16;
    const float* __restrict__ aRow = z1 + (size_t)rowA * (HDIM / 2);
    const bool nval = (l16 < NROLE);
    const float* __restrict__ wRow = wc2 + (size_t)l16 * (HDIM / 2);

    v8f acc = {};
#pragma unroll
    for (int kb = 0; kb < HDIM / 2; kb += 4) {
        const int k = kb + 2 * half;
        v2f a; a.x = aRow[k]; a.y = aRow[k + 1];
        v2f b;
        if (nval) { b.x = wRow[k]; b.y = wRow[k + 1]; }
        else      { b.x = 0.0f;    b.y = 0.0f; }
        acc = wmma_f32(a, b, acc);
    }
    if (nval) {
        const float bias = bc2[l16];
#pragma unroll
        for (int r = 0; r < 8; r++) {
            const int m = m0 + r + 8 * half;
            out[(size_t)m * NROLE + l16] = acc[r] + bias;
        }
    }
}

// ---------------- host launcher ----------------
extern "C" void kernel_launch(void* const* d_in, const int* in_sizes, int n_in,
                              void* d_out, int out_size, void* d_ws, size_t ws_size,
                              hipStream_t stream) {
    const float* x     = (const float*)d_in[0];
    const int*   edge  = (const int*)d_in[1];
    const int*   src   = edge;
    const int*   dst   = edge + N_EDGES;
    const int*   uids  = (const int*)d_in[2];
    const int*   roles = (const int*)d_in[3];
    const float* w_l0  = (const float*)d_in[4];
    const float* b_l0  = (const float*)d_in[5];
    const float* w_r0  = (const float*)d_in[6];
    const float* w_l1  = (const float*)d_in[7];
    const float* b_l1  = (const float*)d_in[8];
    const float* w_r1  = (const float*)d_in[9];
    const float* w_ih  = (const float*)d_in[10];
    /* w_hh d_in[11] unused: h0 == 0 */
    const float* b_ih  = (const float*)d_in[12];
    const float* b_hh  = (const float*)d_in[13];
    const float* wc0   = (const float*)d_in[14];
    const float* bc0   = (const float*)d_in[15];
    const float* wc1   = (const float*)d_in[16];
    const float* bc1   = (const float*)d_in[17];
    const float* wc2   = (const float*)d_in[18];
    const float* bc2   = (const float*)d_in[19];

    // workspace layout (total ~111.3 MB, regions reused across phases)
    char* ws = (char*)d_ws;
    float* deg   = (float*)ws;                                    //   400 KB (pad 512 KB)
    float* msg   = (float*)(ws + 524288);                         //  51.2 MB
    float* h     = (float*)(ws + 524288 + 51200000);              //  51.2 MB
    float* ue    = (float*)(ws + 524288 + 2ull * 51200000);       //   8.4 MB
    float* gates = msg;                 // reuse: msg free after gather-GEMM (33.5 MB)
    float* lstm  = h;                   // reuse: h free after gather-GEMM (8.4 MB)
    float* z0    = h + BATCH * HDIM;    //   8.4 MB
    float* z1    = z0 + BATCH * HDIM;   //   4.2 MB

    // phase 0: degree (shared by both layers)
    zero_kernel<<<512, 256, 0, stream>>>(deg, N_NODES);
    degree_kernel<<<(N_EDGES + 255) / 256, 256, 0, stream>>>(dst, deg, N_EDGES);

    // phase 1: SAGE layer 0 -> h = relu(mean(x) @ w_l0^T + b_l0 + x @ w_r0^T)
    zero_kernel<<<8192, 256, 0, stream>>>(msg, N_NODES * FDIM);
    scatter_kernel<<<(N_EDGES * 32) / 256, 256, 0, stream>>>(x, src, dst, msg, N_EDGES);
    sage_gemm_kernel<<<N_NODES / 16, 256, 0, stream>>>(msg, deg, x, w_l0, w_r0, b_l0, h, 1);

    // phase 2: SAGE layer 1 aggregation over all edges, GEMM only at user rows -> ue (clipped)
    zero_kernel<<<8192, 256, 0, stream>>>(msg, N_NODES * FDIM);
    scatter_kernel<<<(N_EDGES * 32) / 256, 256, 0, stream>>>(h, src, dst, msg, N_EDGES);
    sage_gemm_gather_kernel<<<BATCH / 16, 256, 0, stream>>>(msg, deg, h, uids, w_l1, w_r1, b_l1, ue);

    // phase 3: LSTM single step
    lstm_gates_kernel<<<dim3(BATCH / 16, 4), 256, 0, stream>>>(ue, x, uids, roles, w_ih, b_ih, b_hh, gates);
    lstm_act_kernel<<<(BATCH * HDIM) / 256, 256, 0, stream>>>(gates, lstm);

    // phase 4: classifier head
    mlp0_kernel<<<BATCH / 16, 256, 0, stream>>>(ue, lstm, wc0, bc0, z0);
    mlp1_kernel<<<BATCH / 16, 128, 0, stream>>>(z0, wc1, bc1, z1);
    mlp2_kernel<<<BATCH / 16, 32, 0, stream>>>(z1, wc2, bc2, (float*)d_out);
}